// QuantumKGE_37022618092125
// MI455X (gfx1250) — compile-verified
//
#include <hip/hip_runtime.h>

// ---------------------------------------------------------------------------
// QuantumKGE on MI455X (gfx1250, wave32).
// score(h,r,t) = Re( v(t)^H * U_r * v(h) ),  v(e)=block(s0,ent[e]), U_r=block unitary.
// Pipeline: precompute v(e) [E x 16 cplx], U_r [R x 16x16 cplx], bucket samples
// by relation, then per 16-sample chunk do complex 16x16 matmul with
// V_WMMA_F32_16X16X4_F32 (4 K-slices x 4 real matmuls = 16 wmma / chunk).
// ---------------------------------------------------------------------------

typedef float v2f __attribute__((ext_vector_type(2)));
typedef float v8f __attribute__((ext_vector_type(8)));

// Apply (possibly controlled) Rot gate. cm = control mask (0 = uncontrolled),
// tm = target bit mask. Qubit q <-> bit (3-q) of the flat index (C-order).
__device__ __forceinline__ void apply_gate(float* sr, float* si,
                                           const float* p, int cm, int tm) {
  float phi = p[0], th = p[1], om = p[2];
  float s_, c_, sa, ca, sb, cb;
  sincosf(0.5f * th, &s_, &c_);
  sincosf(0.5f * (phi + om), &sa, &ca);
  sincosf(0.5f * (phi - om), &sb, &cb);
  // Rot = [[e^{-ia}c, -e^{+ib}s], [e^{-ib}s, e^{+ia}c]], a=(phi+om)/2, b=(phi-om)/2
  float m00r =  ca * c_, m00i = -sa * c_;
  float m01r = -cb * s_, m01i = -sb * s_;
  float m10r =  cb * s_, m10i = -sb * s_;
  float m11r =  ca * c_, m11i =  sa * c_;
#pragma unroll
  for (int i0 = 0; i0 < 16; ++i0) {
    if (i0 & tm) continue;                // target bit must be 0
    if ((i0 & cm) != cm) continue;        // control bit(s) must be 1
    int i1 = i0 | tm;
    float x0r = sr[i0], x0i = si[i0], x1r = sr[i1], x1i = si[i1];
    sr[i0] = m00r*x0r - m00i*x0i + m01r*x1r - m01i*x1i;
    si[i0] = m00r*x0i + m00i*x0r + m01r*x1i + m01i*x1r;
    sr[i1] = m10r*x0r - m10i*x0i + m11r*x1r - m11i*x1i;
    si[i1] = m10r*x0i + m10i*x0r + m11r*x1i + m11i*x1r;
  }
}

__device__ __forceinline__ void run_block(float* sr, float* si, const float* P) {
#pragma unroll
  for (int q = 0; q < 4; ++q)
    apply_gate(sr, si, P + q * 3, 0, 8 >> q);
#pragma unroll
  for (int off = 1; off <= 3; ++off) {
#pragma unroll
    for (int q = 0; q < 4; ++q) {
      int t = (q + off) & 3;
      apply_gate(sr, si, P + (off * 4 + q) * 3, 8 >> q, 8 >> t);
    }
  }
}

// v(e) = block(uniform 1/4 state, entity_params[e])
__global__ void k_entity(const float* __restrict__ ep, float2* __restrict__ vtab, int E) {
  int e = blockIdx.x * blockDim.x + threadIdx.x;
  if (e >= E) return;
  float sr[16], si[16];
#pragma unroll
  for (int i = 0; i < 16; ++i) { sr[i] = 0.25f; si[i] = 0.0f; }
  run_block(sr, si, ep + (size_t)e * 48);
#pragma unroll
  for (int i = 0; i < 16; ++i) vtab[(size_t)e * 16 + i] = make_float2(sr[i], si[i]);
}

// U_r columns: thread (r,col) simulates block on basis state e_col.
// Store row-major: utab[r*512 + i*16 + col] (re), +256 offset (im).
__global__ void k_relation(const float* __restrict__ rp, float* __restrict__ utab, int R) {
  int gid = blockIdx.x * blockDim.x + threadIdx.x;
  if (gid >= R * 16) return;
  int r = gid >> 4, col = gid & 15;
  float sr[16], si[16];
#pragma unroll
  for (int i = 0; i < 16; ++i) { sr[i] = (i == col) ? 1.0f : 0.0f; si[i] = 0.0f; }
  run_block(sr, si, rp + (size_t)r * 48);
  float* ure = utab + (size_t)r * 512;
  float* uim = ure + 256;
#pragma unroll
  for (int i = 0; i < 16; ++i) { ure[i * 16 + col] = sr[i]; uim[i * 16 + col] = si[i]; }
}

__global__ void k_zero(int* __restrict__ cnt, int R) {
  int i = blockIdx.x * blockDim.x + threadIdx.x;
  if (i < R) cnt[i] = 0;
}

__global__ void k_count(const int* __restrict__ r, int* __restrict__ cnt, int B) {
  int i = blockIdx.x * blockDim.x + threadIdx.x;
  if (i < B) atomicAdd(&cnt[r[i]], 1);
}

__global__ void k_scan(const int* __restrict__ cnt, int* __restrict__ offs,
                       int* __restrict__ cursor, int R) {
  if (blockIdx.x == 0 && threadIdx.x == 0) {
    int acc = 0;
    for (int q = 0; q < R; ++q) { offs[q] = acc; cursor[q] = acc; acc += cnt[q]; }
    offs[R] = acc;
  }
}

__global__ void k_scatter(const int* __restrict__ r, int* __restrict__ cursor,
                          int* __restrict__ perm, int B) {
  int i = blockIdx.x * blockDim.x + threadIdx.x;
  if (i < B) { int pos = atomicAdd(&cursor[r[i]], 1); perm[pos] = i; }
}

// One block (8 waves) per relation; each wave scores 16-sample chunks with WMMA.
// A[j][k] = v(h_j)[k], B[k][i] = U_r[i][k]  =>  P[j][i] = (U_r v(h_j))[i].
// f32 16x16x4 operand layout (lanes 0-15 half / 16-31 half, 2 VGPRs = k, k+1
// with k = 4*slice + 2*(lane>=16)) per CDNA5 ISA 7.12.2.
__global__ void __launch_bounds__(256) k_score(
    const float2* __restrict__ vtab, const float* __restrict__ utab,
    const int* __restrict__ h, const int* __restrict__ t,
    const int* __restrict__ offs, const int* __restrict__ cnt,
    const int* __restrict__ perm, float* __restrict__ out) {
  int rr = blockIdx.x;
  int n = cnt[rr];
  if (n == 0) return;
  int base = offs[rr];
  int lane = threadIdx.x & 31;
  int wave = threadIdx.x >> 5;
  int j  = lane & 15;       // sample slot (M) for A, column (N=i) for B
  int hi = lane >> 4;       // lower/upper lane half -> K sub-pair

  __shared__ float lds[8 * 512];          // per-wave: 256 Pr + 256 Pi
  float* Pr = &lds[wave * 512];
  float* Pi = Pr + 256;

  // B = U_r^T, loaded once per wave; bin = -Bi for the Re-part accumulation.
  const float* ure = utab + (size_t)rr * 512;
  const float* uim = ure + 256;
  v2f br[4], bi[4], bin[4];
#pragma unroll
  for (int s2 = 0; s2 < 4; ++s2) {
    int k0 = 4 * s2 + 2 * hi;
    float2 brv = *(const float2*)(ure + j * 16 + k0);   // U_r[i=j][k0], [k0+1]
    float2 biv = *(const float2*)(uim + j * 16 + k0);
    br[s2].x = brv.x;  br[s2].y = brv.y;
    bi[s2].x = biv.x;  bi[s2].y = biv.y;
    bin[s2].x = -biv.x; bin[s2].y = -biv.y;
  }

  int nchunks = (n + 15) >> 4;
  for (int ck = wave; ck < nchunks; ck += 8) {
    int cstart = base + ck * 16;
    int m = n - ck * 16; if (m > 16) m = 16;
    int pj = cstart + (j < m ? j : m - 1);              // clamp idle slots
    int sidx = perm[pj];
    const float4* vh4 = (const float4*)(vtab + (size_t)h[sidx] * 16);

    v8f accR = {}; v8f accI = {};
#pragma unroll
    for (int s2 = 0; s2 < 4; ++s2) {
      int k0 = 4 * s2 + 2 * hi;
      float4 aa = vh4[k0 >> 1];                         // (re0,im0,re1,im1)
      v2f ar; ar.x = aa.x; ar.y = aa.z;
      v2f ai; ai.x = aa.y; ai.y = aa.w;
      // Pr += Ar*Br + Ai*(-Bi);  Pi += Ar*Bi + Ai*Br
      accR = __builtin_amdgcn_wmma_f32_16x16x4_f32(false, ar, false, br[s2],  (short)0, accR, false, false);
      accR = __builtin_amdgcn_wmma_f32_16x16x4_f32(false, ai, false, bin[s2], (short)0, accR, false, false);
      accI = __builtin_amdgcn_wmma_f32_16x16x4_f32(false, ar, false, bi[s2],  (short)0, accI, false, false);
      accI = __builtin_amdgcn_wmma_f32_16x16x4_f32(false, ai, false, br[s2],  (short)0, accI, false, false);
    }

    // D layout: VGPR v holds row M = v + 8*(lane>=16), col N = lane&15.
#pragma unroll
    for (int v = 0; v < 8; ++v) {
      int row = v + hi * 8;
      Pr[row * 16 + j] = accR[v];
      Pi[row * 16 + j] = accI[v];
    }
    asm volatile("s_wait_dscnt 0" ::: "memory");        // same-wave LDS RAW

    if (lane < m) {                                     // lanes 0..15: one sample each
      const float2* vt = vtab + (size_t)t[sidx] * 16;
      float sc = 0.0f;
#pragma unroll
      for (int i = 0; i < 16; ++i) {
        float2 w = vt[i];
        sc += w.x * Pr[lane * 16 + i] + w.y * Pi[lane * 16 + i];
      }
      out[sidx] = sc;
    }
    asm volatile("s_wait_dscnt 0" ::: "memory");
  }
}

extern "C" void kernel_launch(void* const* d_in, const int* in_sizes, int n_in,
                              void* d_out, int out_size, void* d_ws, size_t ws_size,
                              hipStream_t stream) {
  const float* ep = (const float*)d_in[0];
  const float* rp = (const float*)d_in[1];
  const int*   h  = (const int*)d_in[2];
  const int*   r  = (const int*)d_in[3];
  const int*   t  = (const int*)d_in[4];
  int E = in_sizes[0] / 48;
  int R = in_sizes[1] / 48;
  int B = in_sizes[2];
  float* out = (float*)d_out;

  // Workspace carve-up (~15.4 MB total).
  char* ws = (char*)d_ws;
  size_t o = 0;
  float2* vtab = (float2*)(ws + o); o += (size_t)E * 16 * sizeof(float2);
  float*  utab = (float*)(ws + o);  o += (size_t)R * 512 * sizeof(float);
  int* cnt    = (int*)(ws + o);     o += (size_t)R * sizeof(int);
  int* offs   = (int*)(ws + o);     o += (size_t)(R + 1) * sizeof(int);
  int* cursor = (int*)(ws + o);     o += (size_t)R * sizeof(int);
  int* perm   = (int*)(ws + o);     o += (size_t)B * sizeof(int);
  (void)n_in; (void)ws_size; (void)out_size;

  k_entity  <<<(E + 255) / 256, 256, 0, stream>>>(ep, vtab, E);
  k_relation<<<(R * 16 + 255) / 256, 256, 0, stream>>>(rp, utab, R);
  k_zero    <<<(R + 255) / 256, 256, 0, stream>>>(cnt, R);
  k_count   <<<(B + 255) / 256, 256, 0, stream>>>(r, cnt, B);
  k_scan    <<<1, 1, 0, stream>>>(cnt, offs, cursor, R);
  k_scatter <<<(B + 255) / 256, 256, 0, stream>>>(r, cursor, perm, B);
  k_score   <<<R, 256, 0, stream>>>(vtab, utab, h, t, offs, cnt, perm, out);
}